// S4D_model_24292335026634
// MI455X (gfx1250) — compile-verified
//
#include <hip/hip_runtime.h>
#include <hip/hip_bf16.h>
#include <math.h>

typedef __attribute__((ext_vector_type(2))) float v2f;
typedef __attribute__((ext_vector_type(8))) float v8f;

#define BB   16
#define LL   2048
#define HH   512
#define NN2  32
#define LC   64
#define NCH  32            // LL / LC
#define NCOL 512           // NCH * BB  (columns per h; col = c*16 + b)

// workspace layout (float offsets)
#define OFF_W    0                         // w (complex, H*N2)        32768 f
#define OFF_CN   (OFF_W   + 2*HH*NN2)      // Cn = C*dtB (complex)     32768 f
#define OFF_WLC  (OFF_CN  + 2*HH*NN2)      // w^64 (complex)           32768 f
#define OFF_K    (OFF_WLC + 2*HH*NN2)      // K taps (H x 64)          32768 f
#define OFF_E    (OFF_K   + HH*LC)         // E tables (H x 64 x 64)   2M f
#define OFF_G    (OFF_E   + HH*LC*LC)      // G tables (H x 64 x 64)   2M f
#define OFF_T    (OFF_G   + HH*LC*LC)      // Toeplitz (H x 64 x 64)   2M f
#define OFF_Z    (OFF_T   + HH*LC*LC)      // Z / S states (H x 64 x 512) 16M f

static __device__ __forceinline__ v8f wmma_f32(v2f a, v2f b, v8f c) {
  return __builtin_amdgcn_wmma_f32_16x16x4_f32(false, a, false, b, (short)0, c,
                                               false, false);
}

// ---- Stage 1: discretized parameters w = dtA, Cn = C*dtB --------------------
__global__ void k_params(const float* __restrict__ log_dt,
                         const float* __restrict__ Bre,
                         const float* __restrict__ Bim,
                         const float* __restrict__ Cc,
                         const float* __restrict__ Areal,
                         const float* __restrict__ Aimag,
                         float* __restrict__ ws) {
  int t = blockIdx.x * blockDim.x + threadIdx.x;
  if (t >= HH * NN2) return;
  int h = t >> 5;
  float dt = expf(log_dt[h]);
  float Ar = -expf(Areal[t]) - 1e-6f;
  float Ai = Aimag[t];
  // denom = 1 - dt*A/2 + 1e-6
  float dr = 1.0f - 0.5f * dt * Ar + 1e-6f;
  float di = -0.5f * dt * Ai;
  float inv = 1.0f / (dr * dr + di * di);
  // w = (1 + dt*A/2) / denom
  float nr = 1.0f + 0.5f * dt * Ar;
  float ni = 0.5f * dt * Ai;
  float wr = (nr * dr + ni * di) * inv;
  float wi = (ni * dr - nr * di) * inv;
  // dtB = dt*(Bre + i*Bim)/denom ; Cn = C * dtB (C real)
  float br = dt * Bre[t], bi = dt * Bim[t];
  float dbr = (br * dr + bi * di) * inv;
  float dbi = (bi * dr - br * di) * inv;
  float cv = Cc[t];
  ((float2*)(ws + OFF_W))[t]  = make_float2(wr, wi);
  ((float2*)(ws + OFF_CN))[t] = make_float2(cv * dbr, cv * dbi);
}

// ---- Stage 2a: power tables E (w^{63-j}) and G (2*Cn*w^{j+1}), w^64 ---------
__global__ void k_tables(float* __restrict__ ws) {
  int t = blockIdx.x * blockDim.x + threadIdx.x;
  if (t >= HH * NN2) return;
  int h = t >> 5, n = t & 31;
  float2 w  = ((const float2*)(ws + OFF_W))[t];
  float2 cn = ((const float2*)(ws + OFF_CN))[t];
  float* E = ws + OFF_E + (size_t)h * LC * LC;
  float* G = ws + OFF_G + (size_t)h * LC * LC;
  float pr = 1.0f, pi = 0.0f;             // w^0
  for (int j = 0; j < LC; ++j) {
    E[n * LC + (LC - 1 - j)]         = pr;   // Re(w^{63-jj}) at jj = 63-j
    E[(NN2 + n) * LC + (LC - 1 - j)] = pi;   // Im
    float qr = pr * w.x - pi * w.y;          // w^{j+1}
    float qi = pr * w.y + pi * w.x;
    G[j * LC + n]       =  2.0f * (cn.x * qr - cn.y * qi);  //  2*Re(Cn*w^{j+1})
    G[j * LC + NN2 + n] = -2.0f * (cn.x * qi + cn.y * qr);  // -2*Im(Cn*w^{j+1})
    pr = qr; pi = qi;
  }
  ((float2*)(ws + OFF_WLC))[t] = make_float2(pr, pi);       // w^64
}

// ---- Stage 2b: conv taps K[h,j] = 2*Re(sum_n Cn*w^j) ------------------------
__global__ void k_ktap(float* __restrict__ ws) {
  int t = blockIdx.x * blockDim.x + threadIdx.x;
  if (t >= HH * LC) return;
  int h = t >> 6, j = t & 63;
  float s = 0.0f;
  if (j == 0) {
    const float2* CN = (const float2*)(ws + OFF_CN) + h * NN2;
    for (int n = 0; n < NN2; ++n) s += CN[n].x;
    s *= 2.0f;
  } else {
    const float* G = ws + OFF_G + (size_t)h * LC * LC + (j - 1) * LC;
    for (int n = 0; n < NN2; ++n) s += G[n];  // 2*Re(Cn*w^j)
  }
  ws[OFF_K + t] = s;
}

// ---- Stage 2c: lower-triangular Toeplitz T[h][j][m] = K[h][j-m] -------------
__global__ void k_toep(float* __restrict__ ws) {
  int t = blockIdx.x * blockDim.x + threadIdx.x;
  if (t >= HH * LC * LC) return;
  int h = t >> 12, j = (t >> 6) & 63, m = t & 63;
  ws[OFF_T + t] = (j >= m) ? ws[OFF_K + h * LC + (j - m)] : 0.0f;
}

// ---- Stage 3: Z[h][k][col] = sum_j E[h][k][j] * u[b, c*64+j, h]  (WMMA) -----
__global__ __launch_bounds__(32) void k_zgemm(const float* __restrict__ u,
                                              float* __restrict__ ws) {
  int h = blockIdx.y, ct = blockIdx.x;
  int lane = threadIdx.x, lo = lane & 15, hi = lane >> 4;
  int b = lo, c = ct;                      // col = c*16 + b
  const float* E = ws + OFF_E + (size_t)h * LC * LC;
  float* Z = ws + OFF_Z + (size_t)h * LC * NCOL;
  v8f zero = {0.f, 0.f, 0.f, 0.f, 0.f, 0.f, 0.f, 0.f};
  v8f acc[4] = {zero, zero, zero, zero};
#pragma unroll
  for (int k0 = 0; k0 < LC; k0 += 4) {
    int m = k0 + 2 * hi;
    size_t ub = ((size_t)(b * LL + c * LC + m)) * HH + h;
    v2f bu; bu.x = u[ub]; bu.y = u[ub + HH];
#pragma unroll
    for (int mt = 0; mt < 4; ++mt) {
      const float* ap = E + (mt * 16 + lo) * LC + m;
      v2f ae; ae.x = ap[0]; ae.y = ap[1];
      acc[mt] = wmma_f32(ae, bu, acc[mt]);
    }
  }
  int col = ct * 16 + lo;
#pragma unroll
  for (int mt = 0; mt < 4; ++mt)
#pragma unroll
    for (int v = 0; v < 8; ++v)
      Z[(mt * 16 + v + 8 * hi) * NCOL + col] = acc[mt][v];
}

// ---- Stage 3b: chunk-wise scan  s[c+1] = w^64*s[c] + Z[c]  (in place) -------
__global__ void k_scan(float* __restrict__ ws) {
  int t = blockIdx.x * blockDim.x + threadIdx.x;
  if (t >= HH * NN2 * BB) return;
  int b = t & 15, n = (t >> 4) & 31, h = t >> 9;
  float2 wl = ((const float2*)(ws + OFF_WLC))[h * NN2 + n];
  float* Z = ws + OFF_Z + (size_t)h * LC * NCOL;
  float sr = 0.0f, si = 0.0f;
  for (int c = 0; c < NCH; ++c) {
    int col = c * 16 + b;
    float* pr = &Z[n * NCOL + col];
    float* pi = &Z[(NN2 + n) * NCOL + col];
    float zr = *pr, zi = *pi;
    *pr = sr; *pi = si;                      // state at chunk start
    float nsr = wl.x * sr - wl.y * si + zr;
    float nsi = wl.x * si + wl.y * sr + zi;
    sr = nsr; si = nsi;
  }
}

// ---- Stage 4: Y = T@U + G@S, epilogue gelu(Y + D*u)  (WMMA, fused) ----------
__global__ __launch_bounds__(32) void k_main(const float* __restrict__ u,
                                             const float* __restrict__ Dp,
                                             float* __restrict__ out,
                                             const float* __restrict__ ws) {
  int h = blockIdx.y, ct = blockIdx.x;
  int lane = threadIdx.x, lo = lane & 15, hi = lane >> 4;
  int b = lo, c = ct;
  const float* T = ws + OFF_T + (size_t)h * LC * LC;
  const float* G = ws + OFF_G + (size_t)h * LC * LC;
  const float* S = ws + OFF_Z + (size_t)h * LC * NCOL;
  int col = ct * 16 + lo;
  v8f zero = {0.f, 0.f, 0.f, 0.f, 0.f, 0.f, 0.f, 0.f};
  v8f acc[4] = {zero, zero, zero, zero};
#pragma unroll
  for (int k0 = 0; k0 < LC; k0 += 4) {
    int m = k0 + 2 * hi;
    size_t ub = ((size_t)(b * LL + c * LC + m)) * HH + h;
    v2f bu; bu.x = u[ub]; bu.y = u[ub + HH];
    v2f bs; bs.x = S[m * NCOL + col]; bs.y = S[(m + 1) * NCOL + col];
#pragma unroll
    for (int mt = 0; mt < 4; ++mt) {
      int row = (mt * 16 + lo) * LC + m;
      v2f at; at.x = T[row]; at.y = T[row + 1];
      v2f ag; ag.x = G[row]; ag.y = G[row + 1];
      acc[mt] = wmma_f32(at, bu, acc[mt]);   // intra-chunk Toeplitz conv
      acc[mt] = wmma_f32(ag, bs, acc[mt]);   // inter-chunk state readout
    }
  }
  float Dv = Dp[0];
#pragma unroll
  for (int mt = 0; mt < 4; ++mt) {
#pragma unroll
    for (int v = 0; v < 8; ++v) {
      int j = mt * 16 + v + 8 * hi;
      size_t idx = ((size_t)(b * LL + c * LC + j)) * HH + h;
      float x = acc[mt][v] + Dv * u[idx];
      out[idx] = 0.5f * x * (1.0f + erff(x * 0.70710678118654752f)); // exact gelu
    }
  }
}

extern "C" void kernel_launch(void* const* d_in, const int* in_sizes, int n_in,
                              void* d_out, int out_size, void* d_ws,
                              size_t ws_size, hipStream_t stream) {
  (void)in_sizes; (void)n_in; (void)out_size; (void)ws_size;
  const float* u      = (const float*)d_in[0];
  const float* log_dt = (const float*)d_in[1];
  const float* Bre    = (const float*)d_in[2];
  const float* Bim    = (const float*)d_in[3];
  const float* Cc     = (const float*)d_in[4];
  const float* Areal  = (const float*)d_in[5];
  const float* Aimag  = (const float*)d_in[6];
  const float* Dp     = (const float*)d_in[7];
  float* ws  = (float*)d_ws;
  float* out = (float*)d_out;

  k_params<<<(HH * NN2 + 255) / 256, 256, 0, stream>>>(log_dt, Bre, Bim, Cc,
                                                       Areal, Aimag, ws);
  k_tables<<<(HH * NN2 + 255) / 256, 256, 0, stream>>>(ws);
  k_ktap<<<(HH * LC + 255) / 256, 256, 0, stream>>>(ws);
  k_toep<<<(HH * LC * LC + 255) / 256, 256, 0, stream>>>(ws);
  k_zgemm<<<dim3(NCOL / 16, HH), 32, 0, stream>>>(u, ws);
  k_scan<<<(HH * NN2 * BB + 255) / 256, 256, 0, stream>>>(ws);
  k_main<<<dim3(NCOL / 16, HH), 32, 0, stream>>>(u, Dp, out, ws);
}